// TwoConvSAGE_50448685859139
// MI455X (gfx1250) — compile-verified
//
#include <hip/hip_runtime.h>
#include <hip/hip_bf16.h>

// ---------------------------------------------------------------------------
// TwoConvSAGE on MI455X (gfx1250), compile-only optimized.
//
// Key algebraic rewrites (exact in real arithmetic):
//   * Linear->Linear with no activation fuses: W1 = W1a@W1b [256,12],
//     bias1 = b1a@W1b + b1b. W2 = W2a@W2b [24,12].
//   * segment_mean is linear: mean_{edges}(h[src]) @ Wbot
//       == mean_{edges}((h @ Wbot)[src]).
//     So project nodes FIRST (50000x12), then scatter 12 floats/edge
//     instead of 128 -> ~10x less edge bandwidth.
//
// Node projections use V_WMMA_F32_16X16X4_F32 (wave32, 16x16 f32 tiles).
// Weights staged in LDS padded to 16 columns so the inner loop has zero
// divergence (no exec save/restore around the ds loads feeding WMMA).
// ---------------------------------------------------------------------------

#define NNODES   50000
#define NEDGES   800000
#define NGRAPHS  512
#define INDIM    128
#define HIDDEN   256
#define MIDD     12
#define NCLS     10
#define NPAD     16        // padded N dimension for WMMA B/C tiles

typedef __attribute__((ext_vector_type(2))) float v2f;
typedef __attribute__((ext_vector_type(8))) float v8f;

// ---- workspace layout (float offsets) -------------------------------------
#define OFF_W1    0         // 256*12 fused layer-1 weights
#define OFF_B1    3072      // 12
#define OFF_W2    3088      // 24*12 fused layer-2 weights
#define OFF_B2    3376      // 12
#define OFF_DEG   3392      // 50000 in-degree (float)
#define OFF_GSUM  53392     // 512*12 graph pooling sums
#define OFF_GCNT  59536     // 512 graph node counts
#define OFF_P0    60064     // 50000*12  pself / qself
#define OFF_P1    660064    // 50000*12  pneigh / qn
#define OFF_P2    1260064   // 50000*12  agg (scatter-mean accumulator)
#define OFF_P3    1860064   // 50000*12  h1
// total: 2,460,064 floats = ~9.8 MB of d_ws

// ---------------------------------------------------------------------------
__global__ void zero_kernel(float* __restrict__ p, int n) {
    int i = blockIdx.x * blockDim.x + threadIdx.x;
    if (i < n) p[i] = 0.0f;
}

// Fuse (W_a @ W_b) for both layers + fold biases. One block of 256 threads.
__global__ void fuse_weights_kernel(const float* __restrict__ W1a,
                                    const float* __restrict__ b1a,
                                    const float* __restrict__ W1b,
                                    const float* __restrict__ b1b,
                                    const float* __restrict__ W2a,
                                    const float* __restrict__ b2a,
                                    const float* __restrict__ W2b,
                                    const float* __restrict__ b2b,
                                    float* __restrict__ ws) {
    __shared__ float l1b[HIDDEN * MIDD];  // 12 KB
    __shared__ float l2b[HIDDEN * MIDD];  // 12 KB
    int tid = threadIdx.x;
    for (int i = tid; i < HIDDEN * MIDD; i += 256) { l1b[i] = W1b[i]; l2b[i] = W2b[i]; }
    __syncthreads();

    // W1 fused: row tid of [256,12]
    {
        float acc[MIDD];
        #pragma unroll
        for (int c = 0; c < MIDD; ++c) acc[c] = 0.0f;
        for (int k = 0; k < HIDDEN; ++k) {
            float a = W1a[tid * HIDDEN + k];
            #pragma unroll
            for (int c = 0; c < MIDD; ++c) acc[c] += a * l1b[k * MIDD + c];
        }
        #pragma unroll
        for (int c = 0; c < MIDD; ++c) ws[OFF_W1 + tid * MIDD + c] = acc[c];
    }
    // W2 fused: rows 0..23 of [24,12]
    if (tid < 2 * MIDD) {
        float acc[MIDD];
        #pragma unroll
        for (int c = 0; c < MIDD; ++c) acc[c] = 0.0f;
        for (int k = 0; k < HIDDEN; ++k) {
            float a = W2a[tid * HIDDEN + k];
            #pragma unroll
            for (int c = 0; c < MIDD; ++c) acc[c] += a * l2b[k * MIDD + c];
        }
        #pragma unroll
        for (int c = 0; c < MIDD; ++c) ws[OFF_W2 + tid * MIDD + c] = acc[c];
    }
    // biases
    if (tid < MIDD) {
        float s1 = 0.0f, s2 = 0.0f;
        for (int k = 0; k < HIDDEN; ++k) {
            s1 += b1a[k] * l1b[k * MIDD + tid];
            s2 += b2a[k] * l2b[k * MIDD + tid];
        }
        ws[OFF_B1 + tid] = s1 + b1b[tid];
        ws[OFF_B2 + tid] = s2 + b2b[tid];
    }
}

// In-degree (same for both layers).
__global__ void degree_kernel(const int* __restrict__ dst, float* __restrict__ deg) {
    int e = blockIdx.x * blockDim.x + threadIdx.x;
    if (e < NEDGES) atomicAdd(&deg[dst[e]], 1.0f);
}

// ---------------------------------------------------------------------------
// WMMA node projection: out_self = X @ W[0:K]   (+bias, via C init)
//                       out_nbr  = X @ W[K:2K]
// X: [NNODES, K] row-major, W: [2K, 12] row-major, staged in LDS padded to
// [2K,16] (cols 12..15 = 0) so every lane loads B unconditionally.
// One wave handles one 16-row tile; V_WMMA_F32_16X16X4_F32 over K/4 steps.
// A layout (16x4 f32): lanes 0-15 -> M=lane, K={k,k+1}; lanes 16-31 -> K={k+2,k+3}.
// B layout (4x16 f32): lanes 0-15 -> N=lane, K={k,k+1}; lanes 16-31 -> K={k+2,k+3}.
// C layout: VGPR r: lanes 0-15 -> (M=r, N=lane); lanes 16-31 -> (M=8+r, N=lane-16).
// ---------------------------------------------------------------------------
template <int K>
__global__ void __launch_bounds__(256)
proj_wmma_kernel(const float* __restrict__ X,
                 const float* __restrict__ Wfused,   // [2K,12]
                 const float* __restrict__ bias,     // [12]
                 float* __restrict__ outSelf,        // [NNODES,12]
                 float* __restrict__ outNbr) {       // [NNODES,12]
    __shared__ float lw[2 * K * NPAD];   // padded: col stride 16, cols 12..15 zero
    __shared__ float lbias[NPAD];
    int tid = threadIdx.x;
    for (int i = tid; i < 2 * K * NPAD; i += 256) {
        const int c = i & (NPAD - 1);
        const int r = i >> 4;
        lw[i] = (c < MIDD) ? Wfused[r * MIDD + c] : 0.0f;
    }
    if (tid < NPAD) lbias[tid] = (tid < MIDD) ? bias[tid] : 0.0f;
    __syncthreads();

    const int wave = tid >> 5;
    const int lane = tid & 31;
    const int half = lane >> 4;   // 0: K pair {0,1}; 1: K pair {2,3}
    const int l16  = lane & 15;

    const int tile = blockIdx.x * 8 + wave;
    const long base = (long)tile * 16;
    if (base >= NNODES) return;            // wave-uniform exit; EXEC stays full

    const int col = l16;                    // N index (only 0..11 stored)
    const float bcol = lbias[col];

    v8f accS, accN;
    #pragma unroll
    for (int r = 0; r < 8; ++r) { accS[r] = bcol; accN[r] = 0.0f; }

    const int arow = (int)base + l16;       // tiles are exact: 50000 = 3125*16
    const float* __restrict__ ap = X + (long)arow * K;
    const float* __restrict__ wS = lw + col;               // self rows [0,K)
    const float* __restrict__ wN = lw + K * NPAD + col;    // neigh rows [K,2K)

    #pragma unroll 4
    for (int k = 0; k < K; k += 4) {
        const int ka = k + half * 2;
        v2f a;
        a.x = ap[ka + 0];
        a.y = ap[ka + 1];

        v2f bS, bN;
        bS.x = wS[(ka + 0) * NPAD];
        bS.y = wS[(ka + 1) * NPAD];
        bN.x = wN[(ka + 0) * NPAD];
        bN.y = wN[(ka + 1) * NPAD];

        accS = __builtin_amdgcn_wmma_f32_16x16x4_f32(false, a, false, bS,
                                                     (short)0, accS, false, false);
        accN = __builtin_amdgcn_wmma_f32_16x16x4_f32(false, a, false, bN,
                                                     (short)0, accN, false, false);
    }

    if (col < MIDD) {
        #pragma unroll
        for (int r = 0; r < 8; ++r) {
            const long m = base + half * 8 + r;
            outSelf[m * MIDD + col] = accS[r];
            outNbr [m * MIDD + col] = accN[r];
        }
    }
}

// Edge scatter-add: agg[dst] += proj[src] (12 floats/edge, float4 vectorized).
__global__ void scatter_kernel(const int* __restrict__ src,
                               const int* __restrict__ dst,
                               const float* __restrict__ proj,   // [N,12]
                               float* __restrict__ agg) {        // [N,12]
    int e = blockIdx.x * blockDim.x + threadIdx.x;
    if (e >= NEDGES) return;
    const long s = src[e], d = dst[e];
    const float4* __restrict__ p = (const float4*)(proj + s * MIDD);
    float4 v0 = p[0], v1 = p[1], v2 = p[2];
    float* __restrict__ a = agg + d * MIDD;
    atomicAdd(a + 0,  v0.x); atomicAdd(a + 1,  v0.y);
    atomicAdd(a + 2,  v0.z); atomicAdd(a + 3,  v0.w);
    atomicAdd(a + 4,  v1.x); atomicAdd(a + 5,  v1.y);
    atomicAdd(a + 6,  v1.z); atomicAdd(a + 7,  v1.w);
    atomicAdd(a + 8,  v2.x); atomicAdd(a + 9,  v2.y);
    atomicAdd(a + 10, v2.z); atomicAdd(a + 11, v2.w);
}

// h = relu(self + (deg>0 ? agg/deg : 0)); element-parallel over N*12.
__global__ void finalize_kernel(const float* __restrict__ pself,
                                const float* __restrict__ agg,
                                const float* __restrict__ deg,
                                float* __restrict__ hout) {
    int i = blockIdx.x * blockDim.x + threadIdx.x;
    if (i >= NNODES * MIDD) return;
    const int n = i / MIDD;
    const float dg = deg[n];
    float v = pself[i] + (dg > 0.0f ? agg[i] / dg : 0.0f);
    hout[i] = v > 0.0f ? v : 0.0f;
}

// Layer-2 finalize fused with graph pooling accumulation.
__global__ void finalize_pool_kernel(const float* __restrict__ qself,
                                     const float* __restrict__ agg,
                                     const float* __restrict__ deg,
                                     const int* __restrict__ graph_id,
                                     float* __restrict__ gsum,   // [G,12]
                                     float* __restrict__ gcnt) { // [G]
    int n = blockIdx.x * blockDim.x + threadIdx.x;
    if (n >= NNODES) return;
    const float dg = deg[n];
    const float inv = dg > 0.0f ? 1.0f / dg : 0.0f;
    const int g = graph_id[n];
    const long base = (long)n * MIDD;
    #pragma unroll
    for (int j = 0; j < MIDD; ++j) {
        float v = qself[base + j] + agg[base + j] * inv;
        v = v > 0.0f ? v : 0.0f;
        atomicAdd(&gsum[(long)g * MIDD + j], v);
    }
    atomicAdd(&gcnt[g], 1.0f);
}

// out[g,c] = mean_graph(h2)[g] @ Wc + bc  (512x10)
__global__ void classify_kernel(const float* __restrict__ gsum,
                                const float* __restrict__ gcnt,
                                const float* __restrict__ Wc,   // [12,10]
                                const float* __restrict__ bc,   // [10]
                                float* __restrict__ out) {      // [512,10]
    int i = blockIdx.x * blockDim.x + threadIdx.x;
    if (i >= NGRAPHS * NCLS) return;
    const int g = i / NCLS, c = i % NCLS;
    const float cnt = gcnt[g];
    const float inv = cnt > 0.0f ? 1.0f / cnt : 0.0f;
    float acc = bc[c];
    #pragma unroll
    for (int j = 0; j < MIDD; ++j)
        acc += (gsum[g * MIDD + j] * inv) * Wc[j * NCLS + c];
    out[i] = acc;
}

// ---------------------------------------------------------------------------
extern "C" void kernel_launch(void* const* d_in, const int* in_sizes, int n_in,
                              void* d_out, int out_size, void* d_ws, size_t ws_size,
                              hipStream_t stream) {
    (void)in_sizes; (void)n_in; (void)out_size; (void)ws_size;
    const float* h   = (const float*)d_in[0];
    const int*   src = (const int*)  d_in[1];
    const int*   dst = (const int*)  d_in[2];
    const int*   gid = (const int*)  d_in[3];
    const float* W1a = (const float*)d_in[4];
    const float* b1a = (const float*)d_in[5];
    const float* W1b = (const float*)d_in[6];
    const float* b1b = (const float*)d_in[7];
    const float* W2a = (const float*)d_in[8];
    const float* b2a = (const float*)d_in[9];
    const float* W2b = (const float*)d_in[10];
    const float* b2b = (const float*)d_in[11];
    const float* Wc  = (const float*)d_in[12];
    const float* bc  = (const float*)d_in[13];
    float* out = (float*)d_out;
    float* ws  = (float*)d_ws;

    const int NM = NNODES * MIDD;                    // 600000
    const int EB = (NEDGES + 255) / 256;             // 3125
    const int PB = ((NNODES / 16) + 7) / 8;          // 391 (3125 tiles, 8 waves/block)
    const int FB = (NM + 255) / 256;                 // 2344

    // zero degree + pooling accumulators + layer-1 agg
    zero_kernel<<<(NNODES + 255) / 256, 256, 0, stream>>>(ws + OFF_DEG, NNODES);
    zero_kernel<<<(6672 + 255) / 256, 256, 0, stream>>>(ws + OFF_GSUM, 6672); // gsum+gcnt(+pad)
    zero_kernel<<<FB, 256, 0, stream>>>(ws + OFF_P2, NM);

    fuse_weights_kernel<<<1, 256, 0, stream>>>(W1a, b1a, W1b, b1b, W2a, b2a, W2b, b2b, ws);
    degree_kernel<<<EB, 256, 0, stream>>>(dst, ws + OFF_DEG);

    // ---- layer 1 ----
    proj_wmma_kernel<INDIM><<<PB, 256, 0, stream>>>(h, ws + OFF_W1, ws + OFF_B1,
                                                    ws + OFF_P0, ws + OFF_P1);
    scatter_kernel<<<EB, 256, 0, stream>>>(src, dst, ws + OFF_P1, ws + OFF_P2);
    finalize_kernel<<<FB, 256, 0, stream>>>(ws + OFF_P0, ws + OFF_P2, ws + OFF_DEG, ws + OFF_P3);

    // ---- layer 2 ----
    zero_kernel<<<FB, 256, 0, stream>>>(ws + OFF_P2, NM);  // reuse agg buffer
    proj_wmma_kernel<MIDD><<<PB, 256, 0, stream>>>(ws + OFF_P3, ws + OFF_W2, ws + OFF_B2,
                                                   ws + OFF_P0, ws + OFF_P1);
    scatter_kernel<<<EB, 256, 0, stream>>>(src, dst, ws + OFF_P1, ws + OFF_P2);
    finalize_pool_kernel<<<(NNODES + 255) / 256, 256, 0, stream>>>(
        ws + OFF_P0, ws + OFF_P2, ws + OFF_DEG, gid, ws + OFF_GSUM, ws + OFF_GCNT);

    // ---- readout ----
    classify_kernel<<<(NGRAPHS * NCLS + 255) / 256, 256, 0, stream>>>(
        ws + OFF_GSUM, ws + OFF_GCNT, Wc, bc, out);
}